// MMG_455266533854
// MI455X (gfx1250) — compile-verified
//
#include <hip/hip_runtime.h>
#include <hip/hip_bf16.h>

typedef float v2f __attribute__((ext_vector_type(2)));
typedef float v8f __attribute__((ext_vector_type(8)));

#define NODES 768
#define FIN   64
#define DEDIM 32
#define DIN   160   // 2*FIN + DEDIM
#define HMID  64
#define HOUT  64
#define FSTR  164   // feat LDS row stride (floats); 16B aligned, spreads banks
#define HSTR  68    // H LDS row stride

// ---------------------------------------------------------------------------
// Phase 0a: compact per-row neighbor lists (order-preserving, wave32 ballot)
// ---------------------------------------------------------------------------
__global__ __launch_bounds__(32)
void build_nbr_kernel(const float* __restrict__ A,
                      int* __restrict__ counts,
                      int* __restrict__ nbr) {
  const int i    = blockIdx.x;
  const int lane = threadIdx.x;
  int cnt = 0;
  for (int j0 = 0; j0 < NODES; j0 += 32) {
    const int j = j0 + lane;
    const bool pred = A[(size_t)i * NODES + j] != 0.0f;
    const unsigned mask = (unsigned)__ballot(pred);
    if (pred) {
      const int pos = cnt + __popc(mask & ((1u << lane) - 1u));
      nbr[i * NODES + pos] = j;
    }
    cnt += __popc(mask);
  }
  if (lane == 0) counts[i] = cnt;
}

// ---------------------------------------------------------------------------
// Phase 0b: swizzle a [Kdim,64] weight matrix into WMMA B-fragment-major
// layout: out[((nt*(Kdim/8) + kkp)*32 + lane)*4 + c]
//   c=0: K=kkp*8+koff,   c=1: K+1,  c=2: K+4,  c=3: K+5   (koff=(lane>>4)*2)
// so one b128 load per lane feeds two 16x16x4 WMMAs.
// ---------------------------------------------------------------------------
__global__ __launch_bounds__(256)
void swizzle_w_kernel(const float* __restrict__ W, float* __restrict__ out,
                      int Kdim) {
  const int idx = blockIdx.x * blockDim.x + threadIdx.x;
  const int KP = Kdim >> 3;
  const int total = 4 * KP * 32 * 4;
  if (idx >= total) return;
  const int c    = idx & 3;
  const int lane = (idx >> 2) & 31;
  const int rest = idx >> 7;           // nt*KP + kkp
  const int kkp  = rest % KP;
  const int nt   = rest / KP;
  const int koff = (lane >> 4) << 1;
  const int n    = nt * 16 + (lane & 15);
  const int kb   = kkp * 8 + ((c >> 1) << 2) + koff + (c & 1);
  out[idx] = W[kb * 64 + n];
}

// ---------------------------------------------------------------------------
// Phase 1/2: EdgeConvE layer. One block (4 waves) per node i.
// Per 16-edge tile: feat[16,160] -> GEMM1 (WMMA f32 16x16x4) -> relu ->
// H[16,64] -> GEMM2 -> relu -> masked column-sum accumulation.
// ---------------------------------------------------------------------------
__global__ __launch_bounds__(128)
void edgeconv_kernel(const float* __restrict__ x,      // [768, 64]
                     const float* __restrict__ eattr,  // [768, 768, 32]
                     const int*   __restrict__ counts, // [768]
                     const int*   __restrict__ nbr,    // [768, 768]
                     const float* __restrict__ WsA,    // swizzled [4][20][32][4]
                     const float* __restrict__ ba,     // [64]
                     const float* __restrict__ WsB,    // swizzled [4][8][32][4]
                     const float* __restrict__ bb,     // [64]
                     float*       __restrict__ out) {  // [768, 64]
  const int i    = blockIdx.x;
  const int t    = threadIdx.x;
  const int wave = t >> 5;        // 0..3 -> n-tile
  const int lane = t & 31;

  __shared__ __align__(16) float xi[FIN];
  __shared__ __align__(16) float feat[16 * FSTR];
  __shared__ __align__(16) float Hs[16 * HSTR];

  if (t < FIN) xi[t] = x[i * FIN + t];
  __syncthreads();

  const int cnt    = counts[i];
  const int ntiles = (cnt + 15) >> 4;

  const int nloc  = wave * 16 + (lane & 15);  // output column 0..63
  const int arow  = lane & 15;                // A-fragment row (M)
  const int koff  = (lane >> 4) << 1;         // lanes 16-31 hold K+2,K+3
  const int rbase = (lane >> 4) << 3;         // C rows: g + 0 or g + 8
  const float biasA = ba[nloc];
  const float biasB = bb[nloc];

  // per-wave fragment streams: one float4 per k-pair per lane
  const float4* __restrict__ fragA =
      (const float4*)WsA + (size_t)wave * (DIN / 8) * 32 + lane;
  const float4* __restrict__ fragB =
      (const float4*)WsB + (size_t)wave * (HMID / 8) * 32 + lane;

  float out_acc = 0.0f;

  for (int tile = 0; tile < ntiles; ++tile) {
    __syncthreads();  // protect previous tile's LDS reads

    // ---- build feat tile: 16 edge slots x 160 features, float4 traffic ---
    {
      const int s    = t >> 3;   // slot 0..15 (8 threads per slot)
      const int sub  = t & 7;
      const int eidx = tile * 16 + s;
      const bool valid = eidx < cnt;
      const int j = valid ? nbr[i * NODES + eidx] : 0;
      const float4* xi4 = (const float4*)xi;
      const float4* xj4 = (const float4*)(x + j * FIN);
      const float4* ej4 = (const float4*)(eattr + ((size_t)i * NODES + j) * DEDIM);
      float4* frow4 = (float4*)(feat + s * FSTR);
#pragma unroll
      for (int r = 0; r < 5; ++r) {
        const int q = sub + (r << 3);  // 0..39 float4 slots
        float4 v = make_float4(0.f, 0.f, 0.f, 0.f);
        if (valid) {
          if (q < 16) {
            v = xi4[q];
          } else if (q < 32) {
            const float4 a = xj4[q - 16];
            const float4 b = xi4[q - 16];
            v = make_float4(a.x - b.x, a.y - b.y, a.z - b.z, a.w - b.w);
          } else {
            v = ej4[q - 32];
          }
        }
        frow4[q] = v;
      }
    }
    __syncthreads();

    // ---- GEMM1: feat[16,160] x Wa[160,64]; 1 b128 weight load / 2 WMMAs --
    v8f acc;
#pragma unroll
    for (int g = 0; g < 8; ++g) acc[g] = biasA;
#pragma unroll 2
    for (int kkp = 0; kkp < DIN / 8; ++kkp) {
      const float4 bw = fragA[kkp * 32];
      const int kb = kkp * 8 + koff;
      v2f a0, a1, b0, b1;
      a0.x = feat[arow * FSTR + kb];
      a0.y = feat[arow * FSTR + kb + 1];
      a1.x = feat[arow * FSTR + kb + 4];
      a1.y = feat[arow * FSTR + kb + 5];
      b0.x = bw.x; b0.y = bw.y;
      b1.x = bw.z; b1.y = bw.w;
      acc = __builtin_amdgcn_wmma_f32_16x16x4_f32(
          false, a0, false, b0, (short)0, acc, false, false);
      acc = __builtin_amdgcn_wmma_f32_16x16x4_f32(
          false, a1, false, b1, (short)0, acc, false, false);
    }
    // relu + store H tile
#pragma unroll
    for (int g = 0; g < 8; ++g) {
      Hs[(g + rbase) * HSTR + nloc] = fmaxf(acc[g], 0.0f);
    }
    __syncthreads();

    // ---- GEMM2: H[16,64] x Wb[64,64] -------------------------------------
    v8f acc2;
#pragma unroll
    for (int g = 0; g < 8; ++g) acc2[g] = biasB;
#pragma unroll 2
    for (int kkp = 0; kkp < HMID / 8; ++kkp) {
      const float4 bw = fragB[kkp * 32];
      const int kb = kkp * 8 + koff;
      v2f a0, a1, b0, b1;
      a0.x = Hs[arow * HSTR + kb];
      a0.y = Hs[arow * HSTR + kb + 1];
      a1.x = Hs[arow * HSTR + kb + 4];
      a1.y = Hs[arow * HSTR + kb + 5];
      b0.x = bw.x; b0.y = bw.y;
      b1.x = bw.z; b1.y = bw.w;
      acc2 = __builtin_amdgcn_wmma_f32_16x16x4_f32(
          false, a0, false, b0, (short)0, acc2, false, false);
      acc2 = __builtin_amdgcn_wmma_f32_16x16x4_f32(
          false, a1, false, b1, (short)0, acc2, false, false);
    }
    // relu, mask invalid edge rows, reduce columns over the 16 rows
    float partial = 0.0f;
#pragma unroll
    for (int g = 0; g < 8; ++g) {
      const int m = g + rbase;
      const float v = fmaxf(acc2[g], 0.0f);
      partial += (tile * 16 + m < cnt) ? v : 0.0f;
    }
    partial += __shfl_xor(partial, 16, 32);  // combine row-halves (lanes l, l+16)
    out_acc += partial;
  }

  if (lane < 16) out[i * HOUT + nloc] = out_acc;
}

// ---------------------------------------------------------------------------
// Phase 3: head  out = sigmoid(relu(x @ W3 + b3) @ W4 + b4)   [768,1]
// ---------------------------------------------------------------------------
__global__ __launch_bounds__(128)
void head_kernel(const float* __restrict__ x,   // [768, 64]
                 const float* __restrict__ W3,  // [64, 128]
                 const float* __restrict__ b3,  // [128]
                 const float* __restrict__ W4,  // [128, 1]
                 const float* __restrict__ b4,  // [1]
                 float* __restrict__ out) {     // [768, 1]
  const int i = blockIdx.x;
  const int t = threadIdx.x;  // 0..127
  __shared__ __align__(16) float xs[64];
  __shared__ float red[128];
  if (t < 64) xs[t] = x[i * 64 + t];
  __syncthreads();
  float h = b3[t];
#pragma unroll 8
  for (int k = 0; k < 64; ++k) h = fmaf(xs[k], W3[k * 128 + t], h);
  h = fmaxf(h, 0.0f);
  red[t] = h * W4[t];
  __syncthreads();
  for (int s2 = 64; s2 > 0; s2 >>= 1) {
    if (t < s2) red[t] += red[t + s2];
    __syncthreads();
  }
  if (t == 0) {
    const float o = red[0] + b4[0];
    out[i] = 1.0f / (1.0f + __expf(-o));
  }
}

// ---------------------------------------------------------------------------
extern "C" void kernel_launch(void* const* d_in, const int* in_sizes, int n_in,
                              void* d_out, int out_size, void* d_ws, size_t ws_size,
                              hipStream_t stream) {
  const float* A   = (const float*)d_in[0];
  const float* x   = (const float*)d_in[1];
  const float* e   = (const float*)d_in[2];
  const float* W1a = (const float*)d_in[3];
  const float* b1a = (const float*)d_in[4];
  const float* W1b = (const float*)d_in[5];
  const float* b1b = (const float*)d_in[6];
  const float* W2a = (const float*)d_in[7];
  const float* b2a = (const float*)d_in[8];
  const float* W2b = (const float*)d_in[9];
  const float* b2b = (const float*)d_in[10];
  const float* W3  = (const float*)d_in[11];
  const float* b3  = (const float*)d_in[12];
  const float* W4  = (const float*)d_in[13];
  const float* b4  = (const float*)d_in[14];

  char* ws = (char*)d_ws;
  size_t off = 0;
  int* counts = (int*)(ws + off);   off += 4096;                        // 768 ints
  int* nbr    = (int*)(ws + off);   off += (size_t)NODES * NODES * 4;   // 2.25 MB
  float* x1   = (float*)(ws + off); off += (size_t)NODES * HOUT * 4;    // 192 KB
  float* x2   = (float*)(ws + off); off += (size_t)NODES * HOUT * 4;    // 192 KB
  float* ws1a = (float*)(ws + off); off += (size_t)DIN  * 64 * 4;       // 40 KB
  float* ws1b = (float*)(ws + off); off += (size_t)HMID * 64 * 4;       // 16 KB
  float* ws2a = (float*)(ws + off); off += (size_t)DIN  * 64 * 4;       // 40 KB
  float* ws2b = (float*)(ws + off); off += (size_t)HMID * 64 * 4;       // 16 KB

  build_nbr_kernel<<<NODES, 32, 0, stream>>>(A, counts, nbr);
  swizzle_w_kernel<<<(DIN  * 64 + 255) / 256, 256, 0, stream>>>(W1a, ws1a, DIN);
  swizzle_w_kernel<<<(HMID * 64 + 255) / 256, 256, 0, stream>>>(W1b, ws1b, HMID);
  swizzle_w_kernel<<<(DIN  * 64 + 255) / 256, 256, 0, stream>>>(W2a, ws2a, DIN);
  swizzle_w_kernel<<<(HMID * 64 + 255) / 256, 256, 0, stream>>>(W2b, ws2b, HMID);

  edgeconv_kernel<<<NODES, 128, 0, stream>>>(x,  e, counts, nbr,
                                             ws1a, b1a, ws1b, b1b, x1);
  edgeconv_kernel<<<NODES, 128, 0, stream>>>(x1, e, counts, nbr,
                                             ws2a, b2a, ws2b, b2b, x2);
  head_kernel<<<NODES, 128, 0, stream>>>(x2, W3, b3, W4, b4, (float*)d_out);
}